// GAT_60842506715222
// MI455X (gfx1250) — compile-verified
//
#include <hip/hip_runtime.h>
#include <hip/hip_bf16.h>

#define N_NODES 50000
#define N_EDGES 800000
#define HID     24
#define HEADS   8
#define EDGE_DIM 8
#define NHID    192            // HEADS*HID
#define M_TILES 3125           // 50000/16 exact
#define N_TILES 12             // 192/16

typedef __attribute__((ext_vector_type(2))) float v2f;
typedef __attribute__((ext_vector_type(8))) float v8f;

// ---------------------------------------------------------------- utilities
__device__ __forceinline__ void atomic_fmax(float* addr, float val) {
    // classic signed/unsigned bit-pun trick; addr initialized to -inf
    if (val >= 0.0f)
        atomicMax((int*)addr, __float_as_int(val));
    else
        atomicMin((unsigned int*)addr, __float_as_uint(val));
}

// ---------------------------------------------------------------- K0: init
__global__ void k_init(float* __restrict__ agg, float* __restrict__ emax,
                       float* __restrict__ denom, float* __restrict__ alpha_sum) {
    long i = (long)blockIdx.x * blockDim.x + threadIdx.x;
    long stride = (long)gridDim.x * blockDim.x;
    const float NEG_INF = __int_as_float(0xff800000);
    for (long j = i; j < (long)N_NODES * NHID; j += stride) agg[j] = 0.0f;
    for (long j = i; j < (long)N_NODES * HEADS; j += stride) {
        emax[j] = NEG_INF;
        denom[j] = 0.0f;
    }
    if (i < HEADS) alpha_sum[i] = 0.0f;
}

// --------------------------------------------- K1: fold att_edge into W_edge
// v_edge[d][h] = sum_c W_edge[d, h*HID + c] * att_edge[h, c]
__global__ void k_vedge(const float* __restrict__ W_edge,
                        const float* __restrict__ att_edge,
                        float* __restrict__ v_edge) {
    int i = threadIdx.x;                 // 64 threads: i = d*HEADS + h
    if (i >= EDGE_DIM * HEADS) return;
    int d = i >> 3, h = i & 7;
    float s = 0.0f;
    #pragma unroll
    for (int c = 0; c < HID; ++c)
        s += W_edge[d * NHID + h * HID + c] * att_edge[h * HID + c];
    v_edge[d * HEADS + h] = s;
}

// ----------------------------------------- K2: xl = x @ W_lin  (WMMA f32 K=4)
// One wave computes one 16x16 output tile; 6 x V_WMMA_F32_16X16X4_F32 over K=24.
__global__ void k_gemm_wmma(const float* __restrict__ x,
                            const float* __restrict__ W,
                            float* __restrict__ xl) {
    int lane = threadIdx.x & 31;
    int wave = (int)((blockIdx.x * blockDim.x + threadIdx.x) >> 5);
    if (wave >= M_TILES * N_TILES) return;          // uniform per wave: EXEC stays full
    int tm = wave / N_TILES;
    int tn = wave % N_TILES;

    int m  = lane & 15;          // A row / C column index
    int hs = lane >> 4;          // half-wave selector
    const float* arow = x + (long)(tm * 16 + m) * HID;
    int bcol = tn * 16 + m;      // for B, lane&15 is the N index

    v8f c = {0.f, 0.f, 0.f, 0.f, 0.f, 0.f, 0.f, 0.f};
    #pragma unroll
    for (int kb = 0; kb < HID; kb += 4) {
        int k0 = kb + hs * 2;
        // A 16x4: lanes 0-15 hold K={0,1}, lanes 16-31 hold K={2,3}
        v2f a; a.x = arow[k0]; a.y = arow[k0 + 1];
        // B 4x16: VGPR0 -> K = hs*2, VGPR1 -> K = hs*2+1, N = lane&15
        v2f b; b.x = W[(long)k0 * NHID + bcol]; b.y = W[(long)(k0 + 1) * NHID + bcol];
        c = __builtin_amdgcn_wmma_f32_16x16x4_f32(
                /*neg_a=*/false, a, /*neg_b=*/false, b,
                /*c_mod=*/(short)0, c, /*reuse_a=*/false, /*reuse_b=*/false);
    }
    // C/D: VGPR r holds row (r + hs*8), col = lane&15
    int n = lane & 15;
    #pragma unroll
    for (int r = 0; r < 8; ++r) {
        int mg = tm * 16 + r + hs * 8;
        xl[(long)mg * NHID + tn * 16 + n] = c[r];
    }
}

// ------------------------------------- K3: a_src / a_dst per (node, head)
__global__ void k_att(const float* __restrict__ xl,
                      const float* __restrict__ att_src,
                      const float* __restrict__ att_dst,
                      float* __restrict__ a_src, float* __restrict__ a_dst) {
    int i = blockIdx.x * blockDim.x + threadIdx.x;   // i = n*HEADS + h
    if (i >= N_NODES * HEADS) return;
    int n = i >> 3, h = i & 7;
    const float* row = xl + (long)n * NHID + h * HID;
    const float* as  = att_src + h * HID;
    const float* ad  = att_dst + h * HID;
    float s = 0.0f, d = 0.0f;
    #pragma unroll
    for (int c = 0; c < HID; ++c) {
        float v = row[c];
        s += v * as[c];
        d += v * ad[c];
    }
    a_src[i] = s;
    a_dst[i] = d;
}

// -------------------------- K4: per-edge logits (leaky relu) + segment max
__global__ void k_edge_logits(const int* __restrict__ src, const int* __restrict__ dst,
                              const float* __restrict__ edge_attr,
                              const float* __restrict__ a_src, const float* __restrict__ a_dst,
                              const float* __restrict__ v_edge,
                              float* __restrict__ ebuf, float* __restrict__ emax) {
    __shared__ float sv[EDGE_DIM * HEADS];
    if (threadIdx.x < EDGE_DIM * HEADS) sv[threadIdx.x] = v_edge[threadIdx.x];
    __syncthreads();
    int e = blockIdx.x * blockDim.x + threadIdx.x;
    if (e >= N_EDGES) return;
    int s = src[e], d = dst[e];
    float ea[EDGE_DIM];
    #pragma unroll
    for (int j = 0; j < EDGE_DIM; ++j) ea[j] = edge_attr[(long)e * EDGE_DIM + j];
    #pragma unroll
    for (int h = 0; h < HEADS; ++h) {
        float ae = 0.0f;
        #pragma unroll
        for (int j = 0; j < EDGE_DIM; ++j) ae += ea[j] * sv[j * HEADS + h];
        float v = a_src[s * HEADS + h] + a_dst[d * HEADS + h] + ae;
        v = (v > 0.0f) ? v : 0.2f * v;               // LeakyReLU(0.2)
        ebuf[(long)e * HEADS + h] = v;
        atomic_fmax(&emax[d * HEADS + h], v);
    }
}

// ------------------------------------ K5: ex = exp(e - emax[dst]); denom +=
__global__ void k_edge_exp(const int* __restrict__ dst,
                           const float* __restrict__ emax,
                           float* __restrict__ ebuf, float* __restrict__ denom) {
    int e = blockIdx.x * blockDim.x + threadIdx.x;
    if (e >= N_EDGES) return;
    int d = dst[e];
    #pragma unroll
    for (int h = 0; h < HEADS; ++h) {
        float v = __expf(ebuf[(long)e * HEADS + h] - emax[d * HEADS + h]);
        ebuf[(long)e * HEADS + h] = v;
        atomicAdd(&denom[d * HEADS + h], v);
    }
}

// ------------- K6: scatter  agg[dst] += xl[src] * alpha   (wave per edge)
__global__ void k_scatter(const int* __restrict__ src, const int* __restrict__ dst,
                          const float* __restrict__ ebuf, const float* __restrict__ denom,
                          const float* __restrict__ xl, float* __restrict__ agg,
                          float* __restrict__ alpha_sum) {
    __shared__ float s_as[HEADS];
    if (threadIdx.x < HEADS) s_as[threadIdx.x] = 0.0f;
    __syncthreads();

    int lane = threadIdx.x & 31;
    int wpb  = blockDim.x >> 5;
    long wave   = (long)blockIdx.x * wpb + (threadIdx.x >> 5);
    long nwaves = (long)gridDim.x * wpb;

    float asum = 0.0f;                       // lanes 0-7 accumulate alpha per head
    for (long e = wave; e < N_EDGES; e += nwaves) {
        int s = src[e], d = dst[e];
        float alpha = 0.0f;
        if (lane < HEADS)
            alpha = ebuf[e * HEADS + lane] / (denom[d * HEADS + lane] + 1e-16f);
        asum += alpha;
        const float* xs = xl + (long)s * NHID;
        float*       ag = agg + (long)d * NHID;
        #pragma unroll
        for (int i = 0; i < NHID / 32; ++i) {
            int c = lane + 32 * i;
            int h = c / HID;
            float ah = __shfl(alpha, h, 32);
            atomicAdd(&ag[c], xs[c] * ah);   // coalesced 128B atomic bursts
        }
    }
    if (lane < HEADS) atomicAdd(&s_as[lane], asum);
    __syncthreads();
    if (threadIdx.x < HEADS) atomicAdd(&alpha_sum[threadIdx.x], s_as[threadIdx.x]);
}

// --------------- K7: head-mean + bias + ELU + W_out dot   (wave per node)
__global__ void k_final(const float* __restrict__ agg, const float* __restrict__ bias,
                        const float* __restrict__ W_out, const float* __restrict__ b_out,
                        float* __restrict__ y) {
    int lane = threadIdx.x & 31;
    int node = (int)((blockIdx.x * blockDim.x + threadIdx.x) >> 5);
    if (node >= N_NODES) return;
    float val = 0.0f;
    if (lane < HID) {
        float s = 0.0f;
        #pragma unroll
        for (int h = 0; h < HEADS; ++h) s += agg[(long)node * NHID + h * HID + lane];
        s = s * (1.0f / HEADS) + bias[lane];
        s = (s > 0.0f) ? s : (__expf(s) - 1.0f);     // ELU
        val = s * W_out[lane];
    }
    #pragma unroll
    for (int off = 16; off > 0; off >>= 1) val += __shfl_down(val, off, 32);
    if (lane == 0) y[node] = val + b_out[0];
}

// ------------------------------------------------ K8: alpha.mean(axis=0)
__global__ void k_alpha_mean(const float* __restrict__ alpha_sum, float* __restrict__ out) {
    int h = threadIdx.x;
    if (h < HEADS) out[h] = alpha_sum[h] * (1.0f / (float)N_EDGES);
}

// ---------------------------------------------------------------- launcher
extern "C" void kernel_launch(void* const* d_in, const int* in_sizes, int n_in,
                              void* d_out, int out_size, void* d_ws, size_t ws_size,
                              hipStream_t stream) {
    const float* x         = (const float*)d_in[0];
    const int*   ei        = (const int*)d_in[1];     // [2, E] int32
    const float* edge_attr = (const float*)d_in[2];
    const float* W_lin     = (const float*)d_in[3];
    const float* att_src   = (const float*)d_in[4];
    const float* att_dst   = (const float*)d_in[5];
    const float* W_edge    = (const float*)d_in[6];
    const float* att_edge  = (const float*)d_in[7];
    const float* bias      = (const float*)d_in[8];
    const float* W_out     = (const float*)d_in[9];
    const float* b_out     = (const float*)d_in[10];
    (void)in_sizes; (void)n_in; (void)out_size; (void)ws_size;

    const int* src = ei;
    const int* dst = ei + N_EDGES;

    // workspace layout (floats)
    float* w        = (float*)d_ws;
    float* xl       = w;                                   // N*192
    float* agg      = xl      + (long)N_NODES * NHID;      // N*192
    float* a_src    = agg     + (long)N_NODES * NHID;      // N*8
    float* a_dst    = a_src   + (long)N_NODES * HEADS;     // N*8
    float* emax     = a_dst   + (long)N_NODES * HEADS;     // N*8
    float* denom    = emax    + (long)N_NODES * HEADS;     // N*8
    float* ebuf     = denom   + (long)N_NODES * HEADS;     // E*8
    float* v_edge   = ebuf    + (long)N_EDGES * HEADS;     // 64
    float* alpha_s  = v_edge  + EDGE_DIM * HEADS;          // 8

    float* y         = (float*)d_out;            // [50000]
    float* alpha_out = y + N_NODES;              // [8]

    k_init<<<2048, 256, 0, stream>>>(agg, emax, denom, alpha_s);
    k_vedge<<<1, 64, 0, stream>>>(W_edge, att_edge, v_edge);

    {   // WMMA GEMM: 3125*12 = 37500 waves, 8 waves/block
        int waves = M_TILES * N_TILES;
        int blocks = (waves + 7) / 8;
        k_gemm_wmma<<<blocks, 256, 0, stream>>>(x, W_lin, xl);
    }
    k_att<<<(N_NODES * HEADS + 255) / 256, 256, 0, stream>>>(xl, att_src, att_dst, a_src, a_dst);
    k_edge_logits<<<(N_EDGES + 255) / 256, 256, 0, stream>>>(src, dst, edge_attr,
                                                             a_src, a_dst, v_edge, ebuf, emax);
    k_edge_exp<<<(N_EDGES + 255) / 256, 256, 0, stream>>>(dst, emax, ebuf, denom);
    k_scatter<<<2048, 256, 0, stream>>>(src, dst, ebuf, denom, xl, agg, alpha_s);
    k_final<<<(N_NODES * 32 + 255) / 256, 256, 0, stream>>>(agg, bias, W_out, b_out, y);
    k_alpha_mean<<<1, 32, 0, stream>>>(alpha_s, alpha_out);
}